// BN1dFitleredMoments_54331336294644
// MI455X (gfx1250) — compile-verified
//
#include <hip/hip_runtime.h>
#include <math.h>

typedef __attribute__((ext_vector_type(2))) float v2f;
typedef __attribute__((ext_vector_type(4))) float v4f;
typedef __attribute__((ext_vector_type(8))) float v8f;

#define THRES_D 4.0
#define ALPHA_D 0.9
#define EPS_D   1e-10

// ---------------------------------------------------------------------------
// Ones-matmul reduction step on the CDNA5 matrix pipe:
//   D[m][n] = sum_k A[m][k]*1 + C[m][n]   =>  sum(D) = sum(C) + 16*sum(A)
// Lane->matrix-element mapping of A is irrelevant for the final tile sum,
// so arbitrary coalesced data can be fed into the A operand.
// ---------------------------------------------------------------------------
static __device__ __forceinline__ v8f wmma_rowsum_acc(v2f a, v2f ones, v8f c) {
  // 8 args: (neg_a, A, neg_b, B, c_mod, C, reuse_a, reuse_b)
  return __builtin_amdgcn_wmma_f32_16x16x4_f32(false, a, false, ones,
                                               (short)0, c, false, false);
}

static __device__ __forceinline__ float wave_reduce_add(float s) {
#pragma unroll
  for (int off = 16; off > 0; off >>= 1) s += __shfl_xor(s, off, 32);
  return s;
}

static __device__ __forceinline__ float tile_sum(v8f c) {
  return ((c[0] + c[1]) + (c[2] + c[3])) + ((c[4] + c[5]) + (c[6] + c[7]));
}

// ---------------------------------------------------------------------------
// Pass 1: per-block partial sum and sum-of-squares via WMMA ones-reduction.
// ---------------------------------------------------------------------------
__global__ __launch_bounds__(256)
void bn_pass1_moments(const float* __restrict__ x, float* __restrict__ partials,
                      long n4, long n) {
  const int tid = threadIdx.x;
  const long gtid   = (long)blockIdx.x * blockDim.x + tid;
  const long stride = (long)gridDim.x * blockDim.x;
  const long full   = n4 / stride;          // unguarded iterations (uniform)

  v2f ones; ones[0] = 1.0f; ones[1] = 1.0f;
  v8f cs = {};   // sum accumulator tile
  v8f cq = {};   // sum-of-squares accumulator tile
  const v4f* x4 = (const v4f*)x;

  // Fast path: every lane fully in-bounds -> no EXEC manipulation in the loop.
  long i = gtid;
  for (long k = 0; k < full; ++k, i += stride) {
    v4f v = __builtin_nontemporal_load(&x4[i]);   // streaming b128 read
    v2f a0; a0[0] = v[0];       a0[1] = v[1];
    v2f a1; a1[0] = v[2];       a1[1] = v[3];
    v2f q0; q0[0] = v[0]*v[0];  q0[1] = v[1]*v[1];
    v2f q1; q1[0] = v[2]*v[2];  q1[1] = v[3]*v[3];
    cs = wmma_rowsum_acc(a0, ones, cs);           // EXEC all-1s
    cs = wmma_rowsum_acc(a1, ones, cs);
    cq = wmma_rowsum_acc(q0, ones, cq);
    cq = wmma_rowsum_acc(q1, ones, cq);
  }

  // Guarded epilogue (uniform outer test; usually skipped for this shape).
  if (full * stride < n4) {
    v4f v = {0.0f, 0.0f, 0.0f, 0.0f};
    if (i < n4) v = __builtin_nontemporal_load(&x4[i]);
    v2f a0; a0[0] = v[0];       a0[1] = v[1];
    v2f a1; a1[0] = v[2];       a1[1] = v[3];
    v2f q0; q0[0] = v[0]*v[0];  q0[1] = v[1]*v[1];
    v2f q1; q1[0] = v[2]*v[2];  q1[1] = v[3]*v[3];
    cs = wmma_rowsum_acc(a0, ones, cs);           // EXEC rejoined to all-1s
    cs = wmma_rowsum_acc(a1, ones, cs);
    cq = wmma_rowsum_acc(q0, ones, cq);
    cq = wmma_rowsum_acc(q1, ones, cq);
  }

  float s = wave_reduce_add(tile_sum(cs));
  float q = wave_reduce_add(tile_sum(cq));

  __shared__ float ls[8], lq[8];
  const int wave = tid >> 5, lane = tid & 31;
  if (lane == 0) { ls[wave] = s; lq[wave] = q; }
  __syncthreads();
  if (tid == 0) {
    float S = 0.0f, Q = 0.0f;
    const int nw = blockDim.x >> 5;
    for (int w = 0; w < nw; ++w) { S += ls[w]; Q += lq[w]; }
    S *= (1.0f / 16.0f);                 // undo the 16x from ones-matmul (exact)
    Q *= (1.0f / 16.0f);
    if (blockIdx.x == 0) {               // scalar tail (n % 4), usually empty
      for (long t = n4 * 4; t < n; ++t) { float e = x[t]; S += e; Q += e * e; }
    }
    partials[2 * (long)blockIdx.x + 0] = S;
    partials[2 * (long)blockIdx.x + 1] = Q;
  }
}

// ---------------------------------------------------------------------------
// Reduce pass-1 partials in f64; emit mask thresholds lo/hi.
// ---------------------------------------------------------------------------
__global__ __launch_bounds__(256)
void bn_reduce1(const float* __restrict__ partials, float* __restrict__ stats,
                int nb, long n) {
  __shared__ double ds[256], dq[256];
  double S = 0.0, Q = 0.0;
  for (int i = threadIdx.x; i < nb; i += blockDim.x) {
    S += (double)partials[2 * (long)i + 0];
    Q += (double)partials[2 * (long)i + 1];
  }
  ds[threadIdx.x] = S; dq[threadIdx.x] = Q;
  __syncthreads();
  for (int off = blockDim.x >> 1; off > 0; off >>= 1) {
    if ((int)threadIdx.x < off) {
      ds[threadIdx.x] += ds[threadIdx.x + off];
      dq[threadIdx.x] += dq[threadIdx.x + off];
    }
    __syncthreads();
  }
  if (threadIdx.x == 0) {
    double mean = ds[0] / (double)n;
    double var  = (dq[0] - ds[0] * mean) / (double)(n - 1);
    double sdev = sqrt(var + EPS_D);
    stats[0] = (float)(mean - THRES_D * sdev);   // lo
    stats[1] = (float)(mean + THRES_D * sdev);   // hi
  }
}

// ---------------------------------------------------------------------------
// Pass 2: masked sum / sum-sq / count via WMMA ones-reduction.
// ---------------------------------------------------------------------------
__global__ __launch_bounds__(256)
void bn_pass2_masked(const float* __restrict__ x, const float* __restrict__ stats,
                     float* __restrict__ partials, long n4, long n) {
  const float lo = stats[0];
  const float hi = stats[1];
  const int tid = threadIdx.x;
  const long gtid   = (long)blockIdx.x * blockDim.x + tid;
  const long stride = (long)gridDim.x * blockDim.x;
  const long full   = n4 / stride;

  v2f ones; ones[0] = 1.0f; ones[1] = 1.0f;
  v8f cs = {};   // masked sum
  v8f cq = {};   // masked sum of squares
  v8f cc = {};   // masked count
  const v4f* x4 = (const v4f*)x;

  long i = gtid;
  for (long k = 0; k < full; ++k, i += stride) {
    v4f v = __builtin_nontemporal_load(&x4[i]);
    float m0 = (v[0] > lo && v[0] < hi) ? 1.0f : 0.0f;
    float m1 = (v[1] > lo && v[1] < hi) ? 1.0f : 0.0f;
    float m2 = (v[2] > lo && v[2] < hi) ? 1.0f : 0.0f;
    float m3 = (v[3] > lo && v[3] < hi) ? 1.0f : 0.0f;
    float e0 = v[0] * m0, e1 = v[1] * m1, e2 = v[2] * m2, e3 = v[3] * m3;
    v2f a0; a0[0] = e0;    a0[1] = e1;
    v2f a1; a1[0] = e2;    a1[1] = e3;
    v2f q0; q0[0] = e0*e0; q0[1] = e1*e1;
    v2f q1; q1[0] = e2*e2; q1[1] = e3*e3;
    v2f c0; c0[0] = m0;    c0[1] = m1;
    v2f c1; c1[0] = m2;    c1[1] = m3;
    cs = wmma_rowsum_acc(a0, ones, cs);
    cs = wmma_rowsum_acc(a1, ones, cs);
    cq = wmma_rowsum_acc(q0, ones, cq);
    cq = wmma_rowsum_acc(q1, ones, cq);
    cc = wmma_rowsum_acc(c0, ones, cc);
    cc = wmma_rowsum_acc(c1, ones, cc);
  }

  if (full * stride < n4) {
    v4f v = {0.0f, 0.0f, 0.0f, 0.0f};
    bool inb = (i < n4);
    if (inb) v = __builtin_nontemporal_load(&x4[i]);
    // masks must be 0 for out-of-bounds lanes (v==0 passes the window test)
    float m0 = (inb && v[0] > lo && v[0] < hi) ? 1.0f : 0.0f;
    float m1 = (inb && v[1] > lo && v[1] < hi) ? 1.0f : 0.0f;
    float m2 = (inb && v[2] > lo && v[2] < hi) ? 1.0f : 0.0f;
    float m3 = (inb && v[3] > lo && v[3] < hi) ? 1.0f : 0.0f;
    float e0 = v[0] * m0, e1 = v[1] * m1, e2 = v[2] * m2, e3 = v[3] * m3;
    v2f a0; a0[0] = e0;    a0[1] = e1;
    v2f a1; a1[0] = e2;    a1[1] = e3;
    v2f q0; q0[0] = e0*e0; q0[1] = e1*e1;
    v2f q1; q1[0] = e2*e2; q1[1] = e3*e3;
    v2f c0; c0[0] = m0;    c0[1] = m1;
    v2f c1; c1[0] = m2;    c1[1] = m3;
    cs = wmma_rowsum_acc(a0, ones, cs);
    cs = wmma_rowsum_acc(a1, ones, cs);
    cq = wmma_rowsum_acc(q0, ones, cq);
    cq = wmma_rowsum_acc(q1, ones, cq);
    cc = wmma_rowsum_acc(c0, ones, cc);
    cc = wmma_rowsum_acc(c1, ones, cc);
  }

  float s = wave_reduce_add(tile_sum(cs));
  float q = wave_reduce_add(tile_sum(cq));
  float c = wave_reduce_add(tile_sum(cc));

  __shared__ float ls[8], lq[8], lc[8];
  const int wave = tid >> 5, lane = tid & 31;
  if (lane == 0) { ls[wave] = s; lq[wave] = q; lc[wave] = c; }
  __syncthreads();
  if (tid == 0) {
    float S = 0.0f, Q = 0.0f, C = 0.0f;
    const int nw = blockDim.x >> 5;
    for (int w = 0; w < nw; ++w) { S += ls[w]; Q += lq[w]; C += lc[w]; }
    S *= (1.0f / 16.0f);
    Q *= (1.0f / 16.0f);
    C *= (1.0f / 16.0f);
    if (blockIdx.x == 0) {
      for (long t = n4 * 4; t < n; ++t) {
        float e = x[t];
        if (e > lo && e < hi) { S += e; Q += e * e; C += 1.0f; }
      }
    }
    partials[3 * (long)blockIdx.x + 0] = S;
    partials[3 * (long)blockIdx.x + 1] = Q;
    partials[3 * (long)blockIdx.x + 2] = C;
  }
}

// ---------------------------------------------------------------------------
// Reduce pass-2 partials in f64; EMA update; emit scale/bias for the affine.
// ---------------------------------------------------------------------------
__global__ __launch_bounds__(256)
void bn_reduce2(const float* __restrict__ partials,
                const float* __restrict__ gamma, const float* __restrict__ beta,
                float* __restrict__ stats, int nb) {
  __shared__ double ds[256], dq[256], dc[256];
  double S = 0.0, Q = 0.0, C = 0.0;
  for (int i = threadIdx.x; i < nb; i += blockDim.x) {
    S += (double)partials[3 * (long)i + 0];
    Q += (double)partials[3 * (long)i + 1];
    C += (double)partials[3 * (long)i + 2];
  }
  ds[threadIdx.x] = S; dq[threadIdx.x] = Q; dc[threadIdx.x] = C;
  __syncthreads();
  for (int off = blockDim.x >> 1; off > 0; off >>= 1) {
    if ((int)threadIdx.x < off) {
      ds[threadIdx.x] += ds[threadIdx.x + off];
      dq[threadIdx.x] += dq[threadIdx.x + off];
      dc[threadIdx.x] += dc[threadIdx.x + off];
    }
    __syncthreads();
  }
  if (threadIdx.x == 0) {
    double cnt      = dc[0];
    double pmean    = ds[0] / cnt;
    double pvar     = (dq[0] - ds[0] * pmean) / (cnt - 1.0);
    double run_mean = (1.0 - ALPHA_D) * pmean;          // ALPHA*0 + (1-a)*pmean
    double run_var  = ALPHA_D * 1.0 + (1.0 - ALPHA_D) * pvar;
    double scale    = (double)gamma[0] / sqrt(run_var + EPS_D);
    double bias     = (double)beta[0] - run_mean * scale;
    stats[2] = (float)scale;
    stats[3] = (float)bias;
  }
}

// ---------------------------------------------------------------------------
// Pass 3: streaming elementwise affine, NT load + NT store.
// ---------------------------------------------------------------------------
__global__ __launch_bounds__(256)
void bn_apply(const float* __restrict__ x, float* __restrict__ out,
              const float* __restrict__ stats, long n4, long n) {
  const float scale = stats[2];
  const float bias  = stats[3];
  const long gtid   = (long)blockIdx.x * blockDim.x + threadIdx.x;
  const long stride = (long)gridDim.x * blockDim.x;
  const long full   = n4 / stride;
  const v4f* x4 = (const v4f*)x;
  v4f* o4 = (v4f*)out;

  long i = gtid;
  for (long k = 0; k < full; ++k, i += stride) {
    v4f v = __builtin_nontemporal_load(&x4[i]);
    v4f o;
    o[0] = fmaf(v[0], scale, bias);
    o[1] = fmaf(v[1], scale, bias);
    o[2] = fmaf(v[2], scale, bias);
    o[3] = fmaf(v[3], scale, bias);
    __builtin_nontemporal_store(o, &o4[i]);
  }
  if (i < n4) {
    v4f v = __builtin_nontemporal_load(&x4[i]);
    v4f o;
    o[0] = fmaf(v[0], scale, bias);
    o[1] = fmaf(v[1], scale, bias);
    o[2] = fmaf(v[2], scale, bias);
    o[3] = fmaf(v[3], scale, bias);
    __builtin_nontemporal_store(o, &o4[i]);
  }
  if (gtid == 0) {
    for (long t = n4 * 4; t < n; ++t) out[t] = fmaf(x[t], scale, bias);
  }
}

// ---------------------------------------------------------------------------
extern "C" void kernel_launch(void* const* d_in, const int* in_sizes, int n_in,
                              void* d_out, int out_size, void* d_ws, size_t ws_size,
                              hipStream_t stream) {
  const float* x     = (const float*)d_in[0];
  const float* gamma = (const float*)d_in[1];
  const float* beta  = (const float*)d_in[2];
  float* out = (float*)d_out;

  const long n  = (long)in_sizes[0];
  const long n4 = n / 4;

  // Workspace layout (floats): [0..15] stats {lo, hi, scale, bias, ...},
  // then NB*2 pass-1 partials, then NB*3 pass-2 partials.
  float* wsf = (float*)d_ws;
  long maxnb = ((long)(ws_size / sizeof(float)) - 16) / 5;
  int NB = 2048;
  if (maxnb < NB) NB = (int)maxnb;
  if (NB < 1) NB = 1;

  float* stats = wsf;
  float* p1 = wsf + 16;
  float* p2 = wsf + 16 + 2L * NB;

  bn_pass1_moments<<<NB, 256, 0, stream>>>(x, p1, n4, n);
  bn_reduce1      <<<1, 256, 0, stream>>>(p1, stats, NB, n);
  bn_pass2_masked <<<NB, 256, 0, stream>>>(x, stats, p2, n4, n);
  bn_reduce2      <<<1, 256, 0, stream>>>(p2, gamma, beta, stats, NB);
  bn_apply        <<<4096, 256, 0, stream>>>(x, out, stats, n4, n);
}